// TopK_970662609131
// MI455X (gfx1250) — compile-verified
//
#include <hip/hip_runtime.h>
#include <math.h>

// Problem constants (match reference setup_inputs)
#define BB 8
#define NN 50000
#define FF 256
#define KK 1024

typedef __attribute__((ext_vector_type(2))) float v2f;
typedef __attribute__((ext_vector_type(8))) float v8f;

// ---------------------------------------------------------------------------
// Kernel 0: inv-norm of scorer (256 floats, one block)
// ---------------------------------------------------------------------------
__global__ void scorer_norm_kernel(const float* __restrict__ scorer,
                                   float* __restrict__ inv_norm) {
    __shared__ float red[256];
    int t = threadIdx.x;
    float v = scorer[t];
    red[t] = v * v;
    __syncthreads();
    for (int s = 128; s > 0; s >>= 1) {
        if (t < s) red[t] += red[t + s];
        __syncthreads();
    }
    if (t == 0) inv_norm[0] = 1.0f / sqrtf(red[0]);
}

// ---------------------------------------------------------------------------
// Kernel 1: scores[b,n] = dot(embs[b,n,:], scorer) * inv_norm + mask[b,n]
// One wave per node: lane loads 8 contiguous f32 (2x float4), shfl-xor reduce.
// Fully coalesced 128B/wave-load streaming of the 410MB embedding tensor.
// ---------------------------------------------------------------------------
__global__ void score_kernel(const float* __restrict__ embs,
                             const float* __restrict__ mask,
                             const float* __restrict__ scorer,
                             const float* __restrict__ inv_norm,
                             float* __restrict__ scores,
                             int total_nodes) {
    int wave = (blockIdx.x * blockDim.x + threadIdx.x) >> 5;
    int lane = threadIdx.x & 31;
    if (wave >= total_nodes) return;

    const float* p = embs + (size_t)wave * FF + lane * 8;
    const float4 e0 = *(const float4*)(p);
    const float4 e1 = *(const float4*)(p + 4);
    const float4 s0 = *(const float4*)(scorer + lane * 8);
    const float4 s1 = *(const float4*)(scorer + lane * 8 + 4);

    float d = e0.x * s0.x + e0.y * s0.y + e0.z * s0.z + e0.w * s0.w
            + e1.x * s1.x + e1.y * s1.y + e1.z * s1.z + e1.w * s1.w;
    // wave32 butterfly reduction
    for (int off = 16; off > 0; off >>= 1) d += __shfl_xor(d, off);

    if (lane == 0) scores[wave] = d * inv_norm[0] + mask[wave];
}

// ---------------------------------------------------------------------------
// Kernel 2: per-batch top-K via 4-pass 8-bit radix select on monotone keys,
// compaction into LDS, then 1024-wide bitonic sort (score desc, idx asc).
// One 1024-thread block per batch; scores are L2-resident (1.6MB total).
// ---------------------------------------------------------------------------
__device__ __forceinline__ unsigned key_of(float s) {
    unsigned u = __float_as_uint(s);
    return (u & 0x80000000u) ? ~u : (u | 0x80000000u);  // monotone: bigger key = bigger float
}

__global__ void __launch_bounds__(1024)
topk_kernel(const float* __restrict__ scores,
            int* __restrict__ topk_idx,
            float* __restrict__ gate) {
    const int b = blockIdx.x;
    const int tid = threadIdx.x;
    const float* sc = scores + (size_t)b * NN;

    __shared__ unsigned hist[256];
    __shared__ unsigned sDigit, sRemain, cntHi, cntEq;
    __shared__ float skey[KK];
    __shared__ int   sidx[KK];

    // ---- radix select: find the K-th largest key ----
    unsigned prefix = 0, pmask = 0, remain = KK;
    for (int shift = 24; shift >= 0; shift -= 8) {
        for (int i = tid; i < 256; i += 1024) hist[i] = 0;
        __syncthreads();
        for (int i = tid; i < NN; i += 1024) {
            unsigned k = key_of(sc[i]);
            if ((k & pmask) == prefix)
                atomicAdd(&hist[(k >> shift) & 255u], 1u);
        }
        __syncthreads();
        if (tid == 0) {
            unsigned cum = 0;
            int d = 255;
            for (; d > 0; --d) {
                unsigned c = hist[d];
                if (cum + c >= remain) break;
                cum += c;
            }
            sDigit = (unsigned)d;
            sRemain = remain - cum;
        }
        __syncthreads();
        prefix |= sDigit << shift;
        pmask  |= 0xFFu << shift;
        remain = sRemain;
        __syncthreads();
    }
    const unsigned T = prefix;          // key of K-th largest
    const unsigned need = remain;       // how many ==T to take
    const unsigned cntG = KK - need;    // count strictly greater than T

    // ---- compaction into LDS ----
    if (tid == 0) { cntHi = 0; cntEq = 0; }
    skey[tid] = -INFINITY;              // safety fill
    sidx[tid] = 0x7FFFFFFF;
    __syncthreads();
    for (int i = tid; i < NN; i += 1024) {
        float s = sc[i];
        unsigned k = key_of(s);
        if (k > T) {
            unsigned p = atomicAdd(&cntHi, 1u);
            skey[p] = s; sidx[p] = i;
        } else if (k == T) {
            unsigned e = atomicAdd(&cntEq, 1u);
            if (e < need) { skey[cntG + e] = s; sidx[cntG + e] = i; }
        }
    }
    __syncthreads();

    // ---- bitonic sort: descending score, ascending index tie-break ----
    for (unsigned ksz = 2; ksz <= KK; ksz <<= 1) {
        for (unsigned j = ksz >> 1; j > 0; j >>= 1) {
            unsigned i = tid, ixj = i ^ j;
            if (ixj > i) {
                bool descBlock = ((i & ksz) == 0);
                float a = skey[i], c = skey[ixj];
                int ia = sidx[i],  ic = sidx[ixj];
                bool aLess = (a < c) || (a == c && ia > ic);
                if (descBlock ? aLess : !aLess) {
                    skey[i] = c; skey[ixj] = a;
                    sidx[i] = ic; sidx[ixj] = ia;
                }
            }
            __syncthreads();
        }
    }

    topk_idx[b * KK + tid] = sidx[tid];
    gate[b * KK + tid] = tanhf(skey[tid]);
}

// ---------------------------------------------------------------------------
// Kernel 3: WMMA epilogue. out[b,f,k] = embs[b, idx[b,k], f] * gate[b,k]
// One wave per 16(f) x 16(k) output tile:
//   D(16x16) = A(16x4) x B(4x16) accumulated over 4 steps (16 nodes total)
//   A[m, j] = embs[idx_j][f0+m]   (feature-major load -> transpose is free)
//   B[j, n] = gate_j * (n == j)   (diagonal gate matrix slice)
// C/D layout stores rows along K -> coalesced 64B stores.
// ---------------------------------------------------------------------------
__global__ void __launch_bounds__(256)
gather_wmma_kernel(const float* __restrict__ embs,
                   const int* __restrict__ topk_idx,
                   const float* __restrict__ gate,
                   float* __restrict__ out) {
    const int lane = threadIdx.x & 31;
    const int w = blockIdx.x * (blockDim.x >> 5) + (threadIdx.x >> 5);
    // tiles per batch: (FF/16)*(KK/16) = 16*64 = 1024
    const int b  = w >> 10;
    const int rem = w & 1023;
    const int kt = rem >> 4;   // 0..63
    const int ft = rem & 15;   // 0..15
    const int f0 = ft << 4;
    const int k0 = kt << 4;

    const int  m  = lane & 15;        // feature row within tile / gate column
    const bool hi = lane >= 16;

    // lane m holds the index/gate for output column (k0+m)
    const int   idxv = topk_idx[b * KK + k0 + m];
    const float gv   = gate[b * KK + k0 + m];

    const float* base = embs + (size_t)b * NN * FF + (f0 + m);

    v8f acc = {};
    #pragma unroll
    for (int t = 0; t < 4; ++t) {
        // A fragment (16x4 f32): VGPR0 = K{0,2}, VGPR1 = K{1,3} across half-waves
        int jA0 = 4 * t + (hi ? 2 : 0);
        int jA1 = jA0 + 1;
        int n0 = __shfl(idxv, jA0);
        int n1 = __shfl(idxv, jA1);
        float a0 = base[(size_t)n0 * FF];     // 64B coalesced per half-wave
        float a1 = base[(size_t)n1 * FF];
        // B fragment (4x16 f32): VGPR0 = rows K{0,1}, VGPR1 = rows K{2,3}
        int r0 = hi ? 1 : 0;
        int r1 = hi ? 3 : 2;
        float b0 = (m == 4 * t + r0) ? gv : 0.0f;
        float b1 = (m == 4 * t + r1) ? gv : 0.0f;

        v2f A = {a0, a1};
        v2f Bm = {b0, b1};
        acc = __builtin_amdgcn_wmma_f32_16x16x4_f32(
            /*neg_a=*/false, A, /*neg_b=*/false, Bm,
            /*c_mod=*/(short)0, acc, /*reuse_a=*/false, /*reuse_b=*/false);
    }

    // D layout: lanes 0-15 -> rows r, cols lane; lanes 16-31 -> rows r+8
    float* outp = out + (size_t)b * FF * KK + (size_t)f0 * KK + k0;
    #pragma unroll
    for (int r = 0; r < 8; ++r) {
        int M = r + (hi ? 8 : 0);
        outp[(size_t)M * KK + m] = acc[r];
    }
}

// ---------------------------------------------------------------------------
extern "C" void kernel_launch(void* const* d_in, const int* in_sizes, int n_in,
                              void* d_out, int out_size, void* d_ws, size_t ws_size,
                              hipStream_t stream) {
    const float* embs   = (const float*)d_in[0];   // [B,N,F] f32
    const float* mask   = (const float*)d_in[1];   // [B,N]   f32
    const float* scorer = (const float*)d_in[2];   // [F,1]   f32
    float* out = (float*)d_out;                    // [B,F,K] f32

    // workspace layout (all 16B aligned)
    float* scores   = (float*)d_ws;                        // B*N floats
    float* gate     = scores + (size_t)BB * NN;            // B*K floats
    int*   topk_idx = (int*)(gate + (size_t)BB * KK);      // B*K ints
    float* inv_norm = (float*)(topk_idx + (size_t)BB * KK);// 1 float

    // 0) scorer inverse norm
    scorer_norm_kernel<<<1, 256, 0, stream>>>(scorer, inv_norm);

    // 1) scores: one wave per node, 400000 waves -> 50000 blocks of 256
    const int total_nodes = BB * NN;
    scorer_norm_kernel; // (no-op ref; ordering via stream)
    score_kernel<<<(total_nodes * 32 + 255) / 256, 256, 0, stream>>>(
        embs, mask, scorer, inv_norm, scores, total_nodes);

    // 2) per-batch top-K select + sort
    topk_kernel<<<BB, 1024, 0, stream>>>(scores, topk_idx, gate);

    // 3) WMMA gather/gate/transpose: 8192 tile-waves, 8 waves per block
    gather_wmma_kernel<<<(BB * (FF / 16) * (KK / 16)) / 8, 256, 0, stream>>>(
        embs, topk_idx, gate, out);
}